// GAT_20091857011528
// MI455X (gfx1250) — compile-verified
//
#include <hip/hip_runtime.h>

// Problem constants (from reference)
constexpr int N_ = 50000;
constexpr int E_ = 800000;
constexpr int F_ = 128;     // = H*D
constexpr int H_ = 8;
constexpr int D_ = 16;
constexpr int C_ = 40;
constexpr int L_ = 2;
constexpr float NEG_SLOPE = 0.2f;
constexpr float LN_EPS = 1e-5f;

typedef __attribute__((ext_vector_type(2))) float v2f;
typedef __attribute__((ext_vector_type(8))) float v8f;

__device__ __forceinline__ float lrelu(float x) { return x > 0.f ? x : NEG_SLOPE * x; }

// sign-aware float atomic max via int/uint atomics (works with -inf init)
__device__ __forceinline__ void atomicMaxF(float* addr, float val) {
  if (val >= 0.f) atomicMax((int*)addr, __float_as_int(val));
  else            atomicMin((unsigned int*)addr, __float_as_uint(val));
}

// ---------------------------------------------------------------------------
// hp[N,128] = X[N,128] @ W[128,128]   (fp32 WMMA, one wave per 16x16 tile)
// block: 256 threads = 8 waves (one per 16-col n-tile); grid: N/16 m-tiles
// ---------------------------------------------------------------------------
__global__ void gemm_node_fc(const float* __restrict__ X,
                             const float* __restrict__ W,
                             float* __restrict__ Y) {
  const int lane = threadIdx.x & 31;
  const int wave = threadIdx.x >> 5;        // n-tile 0..7
  const int m0   = blockIdx.x * 16;
  const int n0   = wave * 16;
  const int half = lane >> 4;               // 0: K+0/1, 1: K+2/3
  const int l16  = lane & 15;
  const int arow = m0 + l16;                // A-frag row (M) for this lane
  const int bcol = n0 + l16;                // B-frag col (N) for this lane

  v8f acc = {};
#pragma unroll
  for (int k = 0; k < F_; k += 4) {
    const int ka = k + 2 * half;
    v2f a;
    a.x = X[arow * F_ + ka];
    a.y = X[arow * F_ + ka + 1];
    v2f b;
    b.x = W[ka * F_ + bcol];
    b.y = W[(ka + 1) * F_ + bcol];
    acc = __builtin_amdgcn_wmma_f32_16x16x4_f32(false, a, false, b,
                                                (short)0, acc, false, false);
  }
#pragma unroll
  for (int r = 0; r < 8; ++r) {
    const int m = m0 + r + 8 * half;        // D layout: VGPR r -> M=r / r+8
    Y[m * F_ + bcol] = acc[r];
  }
}

// ---------------------------------------------------------------------------
// Per (node, head): el/er attention coefficients; init running max/sum.
// ---------------------------------------------------------------------------
__global__ void node_coef(const float* __restrict__ hp,
                          const float* __restrict__ al,   // [H,D] (layer slice)
                          const float* __restrict__ ar,   // [H,D]
                          float* __restrict__ el, float* __restrict__ er,
                          float* __restrict__ mx, float* __restrict__ sm,
                          int total) {
  const int t = blockIdx.x * blockDim.x + threadIdx.x;
  if (t >= total) return;
  const int n = t >> 3;
  const int h = t & 7;
  const float* row = hp + n * F_ + h * D_;
  float sl = 0.f, sr = 0.f;
#pragma unroll
  for (int d = 0; d < D_; ++d) {
    const float x = row[d];
    sl += x * al[h * D_ + d];
    sr += x * ar[h * D_ + d];
  }
  el[t] = sl;
  er[t] = sr;
  mx[t] = __uint_as_float(0xff800000u);     // -inf
  sm[t] = 0.f;
}

__global__ void zero_f32(float* __restrict__ p, int n) {
  const int t = blockIdx.x * blockDim.x + threadIdx.x;
  if (t < n) p[t] = 0.f;
}

// ---------------------------------------------------------------------------
// Edge pass 1: running max per (dst, head)
// ---------------------------------------------------------------------------
__global__ void edge_max(const int* __restrict__ src, const int* __restrict__ dst,
                         const float* __restrict__ el, const float* __restrict__ er,
                         float* __restrict__ mx, int total) {
  const int t = blockIdx.x * blockDim.x + threadIdx.x;
  if (t >= total) return;
  const int e = t >> 3;
  const int h = t & 7;
  const int s = src[e];
  const int d = dst[e];
  const float v = lrelu(el[s * H_ + h] + er[d * H_ + h]);
  atomicMaxF(&mx[d * H_ + h], v);
}

// ---------------------------------------------------------------------------
// Edge pass 2: sum of exp per (dst, head)
// ---------------------------------------------------------------------------
__global__ void edge_sum(const int* __restrict__ src, const int* __restrict__ dst,
                         const float* __restrict__ el, const float* __restrict__ er,
                         const float* __restrict__ mx, float* __restrict__ sm,
                         int total) {
  const int t = blockIdx.x * blockDim.x + threadIdx.x;
  if (t >= total) return;
  const int e = t >> 3;
  const int h = t & 7;
  const int s = src[e];
  const int d = dst[e];
  const float v = lrelu(el[s * H_ + h] + er[d * H_ + h]);
  atomicAdd(&sm[d * H_ + h], __expf(v - mx[d * H_ + h]));
}

// ---------------------------------------------------------------------------
// Edge pass 3: out[dst] += alpha * hp[src]; one wave32 per edge,
// float4 per lane covers all 128 features.
// ---------------------------------------------------------------------------
__global__ void edge_aggregate(const int* __restrict__ src, const int* __restrict__ dst,
                               const float* __restrict__ el, const float* __restrict__ er,
                               const float* __restrict__ mx, const float* __restrict__ sm,
                               const float* __restrict__ hp, float* __restrict__ out) {
  const int e    = blockIdx.x * (blockDim.x >> 5) + (threadIdx.x >> 5);
  const int lane = threadIdx.x & 31;
  if (e >= E_) return;
  const int s = src[e];
  const int d = dst[e];
  const int h = lane >> 2;                  // 4 lanes per head (16 floats)
  const float v     = lrelu(el[s * H_ + h] + er[d * H_ + h]);
  const float alpha = __expf(v - mx[d * H_ + h]) / (sm[d * H_ + h] + 1e-16f);
  const float4 x = *(const float4*)(hp + (size_t)s * F_ + lane * 4);
  float* o = out + (size_t)d * F_ + lane * 4;
  atomicAdd(o + 0, alpha * x.x);
  atomicAdd(o + 1, alpha * x.y);
  atomicAdd(o + 2, alpha * x.z);
  atomicAdd(o + 3, alpha * x.w);
}

// ---------------------------------------------------------------------------
// bias + LayerNorm(128) + ReLU, one wave32 per node, in place (io buffer).
// ---------------------------------------------------------------------------
__global__ void ln_relu(float* __restrict__ io,
                        const float* __restrict__ bias,
                        const float* __restrict__ gamma,
                        const float* __restrict__ beta) {
  const int n    = blockIdx.x * (blockDim.x >> 5) + (threadIdx.x >> 5);
  const int lane = threadIdx.x & 31;
  if (n >= N_) return;
  float4 x = *(const float4*)(io + (size_t)n * F_ + lane * 4);
  const float4 bb = *(const float4*)(bias + lane * 4);
  x.x += bb.x; x.y += bb.y; x.z += bb.z; x.w += bb.w;
  float s  = x.x + x.y + x.z + x.w;
  float s2 = x.x * x.x + x.y * x.y + x.z * x.z + x.w * x.w;
#pragma unroll
  for (int off = 16; off >= 1; off >>= 1) {
    s  += __shfl_xor(s,  off, 32);
    s2 += __shfl_xor(s2, off, 32);
  }
  const float mu  = s * (1.f / 128.f);
  const float var = s2 * (1.f / 128.f) - mu * mu;
  const float inv = rsqrtf(var + LN_EPS);
  const float4 g  = *(const float4*)(gamma + lane * 4);
  const float4 be = *(const float4*)(beta + lane * 4);
  float4 y;
  y.x = fmaxf((x.x - mu) * inv * g.x + be.x, 0.f);
  y.y = fmaxf((x.y - mu) * inv * g.y + be.y, 0.f);
  y.z = fmaxf((x.z - mu) * inv * g.z + be.z, 0.f);
  y.w = fmaxf((x.w - mu) * inv * g.w + be.w, 0.f);
  *(float4*)(io + (size_t)n * F_ + lane * 4) = y;
}

// ---------------------------------------------------------------------------
// out[N,40] = X[N,128] @ Wp[128,40] + bp  (fp32 WMMA, N padded to 48)
// block: 96 threads = 3 waves (3 n-tiles); grid: N/16
// ---------------------------------------------------------------------------
__global__ void gemm_pred(const float* __restrict__ X,
                          const float* __restrict__ Wp,
                          const float* __restrict__ bp,
                          float* __restrict__ Y) {
  const int lane = threadIdx.x & 31;
  const int wave = threadIdx.x >> 5;        // 0..2
  const int m0   = blockIdx.x * 16;
  const int n0   = wave * 16;
  const int half = lane >> 4;
  const int l16  = lane & 15;
  const int arow = m0 + l16;
  const int n    = n0 + l16;
  const bool valid = (n < C_);

  v8f acc = {};
#pragma unroll
  for (int k = 0; k < F_; k += 4) {
    const int ka = k + 2 * half;
    v2f a;
    a.x = X[arow * F_ + ka];
    a.y = X[arow * F_ + ka + 1];
    v2f b;
    b.x = valid ? Wp[ka * C_ + n] : 0.f;
    b.y = valid ? Wp[(ka + 1) * C_ + n] : 0.f;
    acc = __builtin_amdgcn_wmma_f32_16x16x4_f32(false, a, false, b,
                                                (short)0, acc, false, false);
  }
  if (valid) {
    const float bbn = bp[n];
#pragma unroll
    for (int r = 0; r < 8; ++r) {
      const int m = m0 + r + 8 * half;
      Y[m * C_ + n] = acc[r] + bbn;
    }
  }
}

// ---------------------------------------------------------------------------
extern "C" void kernel_launch(void* const* d_in, const int* in_sizes, int n_in,
                              void* d_out, int out_size, void* d_ws, size_t ws_size,
                              hipStream_t stream) {
  const float* feats = (const float*)d_in[0];
  const int*   src   = (const int*)d_in[1];
  const int*   dst   = (const int*)d_in[2];
  const float* Ws    = (const float*)d_in[3];
  const float* al    = (const float*)d_in[4];
  const float* ar    = (const float*)d_in[5];
  const float* bias  = (const float*)d_in[6];
  const float* gamma = (const float*)d_in[7];
  const float* beta  = (const float*)d_in[8];
  const float* Wp    = (const float*)d_in[9];
  const float* bp    = (const float*)d_in[10];
  float* outp = (float*)d_out;

  // workspace layout (floats): hbuf doubles as aggregation target (in-place LN)
  float* ws   = (float*)d_ws;
  float* hbuf = ws;                               // N*128 (layer io / agg)
  float* hp   = hbuf + (size_t)N_ * F_;           // N*128 (fc projection)
  float* el   = hp + (size_t)N_ * F_;             // N*H
  float* er   = el + (size_t)N_ * H_;             // N*H
  float* mx   = er + (size_t)N_ * H_;             // N*H
  float* sm   = mx + (size_t)N_ * H_;             // N*H

  const int NH = N_ * H_;                         // 400000
  const int EH = E_ * H_;                         // 6400000
  const int NF = N_ * F_;                         // 6400000

  for (int i = 0; i < L_; ++i) {
    const float* X = (i == 0) ? feats : hbuf;
    // 1) fc projection (WMMA f32)
    gemm_node_fc<<<N_ / 16, 256, 0, stream>>>(X, Ws + (size_t)i * F_ * F_, hp);
    // 2) attention coefficients + init stats
    node_coef<<<(NH + 255) / 256, 256, 0, stream>>>(
        hp, al + (size_t)i * H_ * D_, ar + (size_t)i * H_ * D_, el, er, mx, sm, NH);
    // 3) zero aggregation buffer (hbuf is free after the GEMM consumed it)
    zero_f32<<<(NF + 255) / 256, 256, 0, stream>>>(hbuf, NF);
    // 4) segment softmax (3 passes) + weighted aggregation
    edge_max<<<(EH + 255) / 256, 256, 0, stream>>>(src, dst, el, er, mx, EH);
    edge_sum<<<(EH + 255) / 256, 256, 0, stream>>>(src, dst, el, er, mx, sm, EH);
    edge_aggregate<<<E_ / 8, 256, 0, stream>>>(src, dst, el, er, mx, sm, hp, hbuf);
    // 5) bias + LayerNorm + ReLU in place -> next layer input
    ln_relu<<<N_ / 8, 256, 0, stream>>>(hbuf, bias + (size_t)i * F_,
                                        gamma + (size_t)i * F_, beta + (size_t)i * F_);
  }
  // final prediction GEMM (WMMA f32, C padded 40 -> 48)
  gemm_pred<<<N_ / 16, 96, 0, stream>>>(hbuf, Wp, bp, outp);
}